// AttentionPooling_31782757990846
// MI455X (gfx1250) — compile-verified
//
#include <hip/hip_runtime.h>
#include <math.h>

#define DD 128
#define NEG_INF (-3.402823466e38f)

typedef __attribute__((ext_vector_type(2))) float v2f;
typedef __attribute__((ext_vector_type(8))) float v8f;

// ---------------- K1: scores = x @ w^T + b  via V_WMMA_F32_16X16X4_F32 ----------
// Per wave: 16 rows. A = x tile (16x4 f32, documented layout), B = w chunk
// replicated across the 16 N columns, so D[m, n] = dot(x[row m], w) for all n.
// C/D layout: VGPR j holds M=j (lanes 0-15) and M=8+j (lanes 16-31).
__global__ __launch_bounds__(256) void k_scores(const float* __restrict__ x,
                                                const float* __restrict__ attw,
                                                const float* __restrict__ attb,
                                                float* __restrict__ scores,
                                                float* __restrict__ blockmax,
                                                int n) {
  const int tid    = threadIdx.x;
  const int lane   = tid & 31;
  const int wave   = tid >> 5;
  const int laneLo = lane & 15;
  const int hi     = (lane >> 4) & 1;          // lanes 16-31 carry K=2,3
  const int r0     = blockIdx.x * 128 + wave * 16;

  int row = r0 + laneLo;
  if (row > n - 1) row = n - 1;                // branchless clamp; EXEC stays full

  const float* xrow = x + (size_t)row * DD + hi * 2;
  const float* wv   = attw + hi * 2;

  v8f c = {};
#pragma unroll
  for (int s = 0; s < 32; ++s) {
    v2f a = *(const v2f*)(xrow + 4 * s);       // A: x[row][k0 + {0,1} or {2,3}]
    v2f b = *(const v2f*)(wv + 4 * s);         // B: w[k0 + {0,1} or {2,3}], all N
    c = __builtin_amdgcn_wmma_f32_16x16x4_f32(false, a, false, b,
                                              (short)0, c, false, false);
  }

  const float bias = attb[0];
  float m8 = NEG_INF;
#pragma unroll
  for (int j = 0; j < 8; ++j) {
    int   orow = r0 + hi * 8 + j;
    float sc   = c[j] + bias;
    if (orow < n) {
      m8 = fmaxf(m8, sc);
      if (laneLo == 0) scores[orow] = sc;      // one writer per (hi, j)
    }
  }

  __shared__ float smax[256];
  smax[tid] = m8;
  __syncthreads();
#pragma unroll
  for (int off = 128; off > 0; off >>= 1) {
    if (tid < off) smax[tid] = fmaxf(smax[tid], smax[tid + off]);
    __syncthreads();
  }
  if (tid == 0) blockmax[blockIdx.x] = smax[0];
}

// ---------------- K2: global max over block maxes; zero gsum --------------------
__global__ __launch_bounds__(256) void k_reduce_max(const float* __restrict__ bm,
                                                    int nb, float* __restrict__ gmax,
                                                    float* __restrict__ gsum) {
  __shared__ float s[256];
  float m = NEG_INF;
  for (int i = threadIdx.x; i < nb; i += 256) m = fmaxf(m, bm[i]);
  s[threadIdx.x] = m;
  __syncthreads();
  for (int off = 128; off > 0; off >>= 1) {
    if (threadIdx.x < off) s[threadIdx.x] = fmaxf(s[threadIdx.x], s[threadIdx.x + off]);
    __syncthreads();
  }
  if (threadIdx.x == 0) { gmax[0] = s[0]; gsum[0] = 0.0f; }
}

// ---------------- K3: p = exp(score - gmax) in place; gsum += sum(p) ------------
__global__ __launch_bounds__(256) void k_expsum(float* __restrict__ scores, int n,
                                                const float* __restrict__ gmax,
                                                float* __restrict__ gsum) {
  const float gm = gmax[0];
  float acc = 0.0f;
  for (int i = blockIdx.x * blockDim.x + threadIdx.x; i < n;
       i += gridDim.x * blockDim.x) {
    float p   = __expf(scores[i] - gm);
    scores[i] = p;
    acc += p;
  }
  __shared__ float s[256];
  s[threadIdx.x] = acc;
  __syncthreads();
  for (int off = 128; off > 0; off >>= 1) {
    if (threadIdx.x < off) s[threadIdx.x] += s[threadIdx.x + off];
    __syncthreads();
  }
  if (threadIdx.x == 0) atomicAdd(gsum, s[0]);
}

// ---------------- K0: zero the output (we accumulate with atomics) --------------
__global__ __launch_bounds__(256) void k_zero(float* __restrict__ p, int n) {
  for (int i = blockIdx.x * blockDim.x + threadIdx.x; i < n;
       i += gridDim.x * blockDim.x)
    p[i] = 0.0f;
}

// ---------------- K5: sorted weighted segment-sum -------------------------------
// 128 threads = one column each; 128 contiguous rows per block. Register
// run-length accumulation, atomic flush at segment boundaries (~2 per block).
__global__ __launch_bounds__(128) void k_segpool(const float* __restrict__ x,
                                                 const long long* __restrict__ batch,
                                                 const float* __restrict__ p,
                                                 const float* __restrict__ gsum,
                                                 float* __restrict__ out, int n) {
  const int   col = threadIdx.x;
  const float inv = 1.0f / gsum[0];
  int base = blockIdx.x * 128;
  if (base >= n) return;
  int end = base + 128;
  if (end > n) end = n;

  long long cur = batch[base];
  float     acc = 0.0f;
  for (int i = base; i < end; ++i) {
    long long sg = batch[i];
    if (sg != cur) {
      atomicAdd(&out[(size_t)cur * DD + col], acc);
      acc = 0.0f;
      cur = sg;
    }
    acc = fmaf(p[i] * inv, x[(size_t)i * DD + col], acc);
  }
  atomicAdd(&out[(size_t)cur * DD + col], acc);
}

extern "C" void kernel_launch(void* const* d_in, const int* in_sizes, int n_in,
                              void* d_out, int out_size, void* d_ws, size_t ws_size,
                              hipStream_t stream) {
  (void)n_in; (void)ws_size;
  const float*     x     = (const float*)d_in[0];
  const long long* batch = (const long long*)d_in[1];
  const float*     attw  = (const float*)d_in[2];
  const float*     attb  = (const float*)d_in[3];
  float*           out   = (float*)d_out;

  const int n   = in_sizes[1];          // N rows (batch has one entry per row)
  const int nb1 = (n + 127) / 128;      // 128 rows per block (8 waves x 16 rows)

  float* scores   = (float*)d_ws;       // [n]
  float* blockmax = scores + n;         // [nb1]
  float* gmax     = blockmax + nb1;     // [1]
  float* gsum     = gmax + 1;           // [1]

  k_zero      <<<2048, 256, 0, stream>>>(out, out_size);
  k_scores    <<<nb1,  256, 0, stream>>>(x, attw, attb, scores, blockmax, n);
  k_reduce_max<<<1,    256, 0, stream>>>(blockmax, nb1, gmax, gsum);
  k_expsum    <<<2048, 256, 0, stream>>>(scores, n, gmax, gsum);
  k_segpool   <<<nb1,  128, 0, stream>>>(x, batch, scores, gsum, out, n);
}